// CoL_Module_59433757442769
// MI455X (gfx1250) — compile-verified
//
#include <hip/hip_runtime.h>

#ifndef __has_builtin
#define __has_builtin(x) 0
#endif

#define IMG_H   128
#define IMG_W   128
#define TILE_H  32
#define LDS_ROWS (TILE_H + 2)          // 34 rows (1 halo row top/bottom)
#define LDS_W   136                    // floats per LDS row: 544 B, 16-B aligned stride
#define COL0    4                      // global col c lives at LDS index c+4 (b128 alignment)
#define NBINS   5
#define NPLANES (32 * 64)

typedef float f4 __attribute__((ext_vector_type(4)));
typedef int   v4i __attribute__((vector_size(16)));
typedef v4i __attribute__((address_space(1))) g_v4i;   // global (AS1) 16-byte chunk
typedef v4i __attribute__((address_space(3))) l_v4i;   // LDS    (AS3) 16-byte chunk

// ---- gfx1250 async global->LDS copy (ASYNCcnt path) ------------------------
__device__ __forceinline__ void async_copy16(const float* g, float* l) {
#if __has_builtin(__builtin_amdgcn_global_load_async_to_lds_b128)
  __builtin_amdgcn_global_load_async_to_lds_b128(
      (g_v4i*)(void*)g,        // (void*) drops const, then generic->AS1 v4i*
      (l_v4i*)(void*)l,        // generic->AS3 v4i*
      0, 0);                   // imm offset, imm cpol
#else
  __attribute__((address_space(3))) float* lp =
      (__attribute__((address_space(3))) float*)(void*)l;
  asm volatile("global_load_async_to_lds_b128 %0, %1, off"
               :: "v"(lp), "v"(g) : "memory");
#endif
}

__device__ __forceinline__ void wait_async0() {
#if __has_builtin(__builtin_amdgcn_s_wait_asynccnt)
  __builtin_amdgcn_s_wait_asynccnt(0);
#else
  asm volatile("s_wait_asynccnt 0" ::: "memory");
#endif
}

// quantize: floor(x*5) clipped to [0,4]; trunc==floor after the clamp
__device__ __forceinline__ int qbin(float x) {
  int b = (int)(x * (float)NBINS);
  return b < 0 ? 0 : (b > NBINS - 1 ? NBINS - 1 : b);
}

// load 6 LDS floats (cols c0-1 .. c0+4 of one tile row) + quantize them
__device__ __forceinline__ void load_row(const float* tilebase, int trow, int c0,
                                         float (&a)[6], int (&b)[6]) {
  const float* rr = tilebase + trow * LDS_W + COL0 + c0;
  a[0] = rr[-1];
  f4 av = *(const f4*)rr;              // conflict-free b128 (lanes tile all 64 banks)
  a[1] = av.x; a[2] = av.y; a[3] = av.z; a[4] = av.w;
  a[5] = rr[4];
  #pragma unroll
  for (int t = 0; t < 6; ++t) b[t] = qbin(a[t]);
}

__global__ __launch_bounds__(256) void col_cooc_kernel(
    const float* __restrict__ x, const float* __restrict__ Wg,
    const float* __restrict__ Lg, float* __restrict__ out)
{
  __shared__ alignas(16) float tile[LDS_ROWS * LDS_W];
  __shared__ float WL[9 * 25];         // fused W[tap]*L[bn][bp] table

  const int tid   = threadIdx.x;
  const int band  = blockIdx.x;                 // 0..3
  const int plane = blockIdx.y;                 // 0..2047 (b*C + c)
  const int h0    = band * TILE_H;
  const float* __restrict__ src = x + (size_t)plane * (IMG_H * IMG_W);

  // build fused table: 25 words per tap -> distinct banks within a tap
  if (tid < 9 * 25) WL[tid] = Wg[tid / 25] * Lg[tid % 25];

  // zero the left/right halo columns (image-boundary cols, always OOB)
  for (int r = tid; r < LDS_ROWS; r += 256) {
    tile[r * LDS_W + (COL0 - 1)]     = 0.0f;
    tile[r * LDS_W + (COL0 + IMG_W)] = 0.0f;
  }

  // async-stage 34 rows x 32 float4 chunks; zero-fill OOB halo rows
  for (int i = tid; i < LDS_ROWS * (IMG_W / 4); i += 256) {
    const int r = i >> 5;
    const int v = i & 31;
    const int g = h0 - 1 + r;
    float* ldst = &tile[r * LDS_W + COL0 + v * 4];
    if ((unsigned)g < (unsigned)IMG_H) {
      async_copy16(src + (size_t)g * IMG_W + (size_t)(v * 4), ldst);
    } else {
      f4 z = {0.f, 0.f, 0.f, 0.f};
      *(f4*)ldst = z;
    }
  }

  wait_async0();
  __syncthreads();

  // each thread: 4-col x 4-row strip; rolling 3-row window so every tile row
  // is loaded + quantized exactly once per strip
  const int c0    = (tid & 31) * 4;    // lanes of a wave cover cols 0..127 contiguously
  const int rbase = (tid >> 5) * 4;    // wave-uniform row base (0,4,...,28)
  float* __restrict__ outp =
      out + (size_t)plane * (IMG_H * IMG_W) + (size_t)(h0 + rbase) * IMG_W + c0;

  float a[3][6];
  int   bq[3][6];
  load_row(tile, rbase + 0, c0, a[0], bq[0]);
  load_row(tile, rbase + 1, c0, a[1], bq[1]);

  #pragma unroll
  for (int k = 0; k < 4; ++k) {
    const int s0 = k % 3;              // window slot for dh=0 (tile row rbase+k)
    const int s1 = (k + 1) % 3;        // dh=1 (center row)
    const int s2 = (k + 2) % 3;        // dh=2
    load_row(tile, rbase + k + 2, c0, a[s2], bq[s2]);

    int bp4[4];
    #pragma unroll
    for (int j = 0; j < 4; ++j) bp4[j] = bq[s1][j + 1] << 2;   // bp in bytes

    f4 acc = {0.f, 0.f, 0.f, 0.f};

    #pragma unroll
    for (int dh = 0; dh < 3; ++dh) {
      const int s = (dh == 0) ? s0 : (dh == 1 ? s1 : s2);
      #pragma unroll
      for (int dw = 0; dw < 3; ++dw) {
        const int tap = dh * 3 + dw;   // compile-time -> folds into ds imm offset
        #pragma unroll
        for (int j = 0; j < 4; ++j) {
          // byte addr = 100*tap (imm) + 20*bn + 4*bp  -> v_mad_u32_u24 + ds_load
          const float lv = *(const float*)((const char*)WL
                              + 100 * tap + 20 * bq[s][j + dw] + bp4[j]);
          acc[j] += lv * a[s][j + dw];
        }
      }
    }

    *(f4*)(outp + (size_t)k * IMG_W) = acc;   // 512B-contiguous per wave
  }
}

extern "C" void kernel_launch(void* const* d_in, const int* in_sizes, int n_in,
                              void* d_out, int out_size, void* d_ws, size_t ws_size,
                              hipStream_t stream) {
  const float* x  = (const float*)d_in[0];   // (32,64,128,128) fp32
  const float* Wg = (const float*)d_in[1];   // (1,3,3) fp32
  const float* Lg = (const float*)d_in[2];   // (5,5) fp32
  float* out = (float*)d_out;

  dim3 grid(IMG_H / TILE_H, NPLANES);        // (4, 2048)
  col_cooc_kernel<<<grid, 256, 0, stream>>>(x, Wg, Lg, out);
}